// ConditionsLayer_72696616452322
// MI455X (gfx1250) — compile-verified
//
#include <hip/hip_runtime.h>
#include <stdint.h>

// out[b,u] = g ? relu(x[b, u%64] - w1[u]) : relu(w2[u] - x[b, u%64]),
//   g = ((u/64) % 2 == 0),  B=1024, D=64, U=8192.  Pure bandwidth problem
// (~33 MB traffic, ~25 MFLOP): stream 128-bit NT stores, stage reused
// operands (x row, w1/w2 slices) into LDS via the CDNA5 Tensor Data Mover.

typedef __attribute__((ext_vector_type(4))) unsigned int v4u;
typedef __attribute__((ext_vector_type(8))) int          v8i;
typedef __attribute__((ext_vector_type(4))) int          v4i;
typedef __attribute__((ext_vector_type(4))) float        v4f;

// Build a TDM descriptor (D#) for a 1D contiguous copy of `nelem` f32 from
// global memory to LDS, and issue TENSOR_LOAD_TO_LDS.
// Group layouts per CDNA5 ISA §8.3-8.6.
__device__ __forceinline__ void tdm_load_1d_f32(const void* gptr,
                                                uint32_t lds_byte_off,
                                                uint32_t nelem) {
  const uint64_t ga = (uint64_t)(uintptr_t)gptr;

  v4u g0;
  g0.x = 1u;                                   // count=1 (valid), user mode, no gather
  g0.y = lds_byte_off;                         // lds_addr [63:32]
  g0.z = (uint32_t)ga;                         // global_addr [95:64]
  g0.w = (uint32_t)((ga >> 32) & 0x01FFFFFFull) // global_addr [120:96]
       | (2u << 30);                           // type=2 ("image") [127:126]

  v8i g1;
  g1[0] = (int)(2u << 16);                     // workgroup_mask=0, data_size=2 (4B)
  g1[1] = (int)((nelem & 0xFFFFu) << 16);      // tensor_dim0[15:0] at bits 63:48
  g1[2] = (int)(((nelem >> 16) & 0xFFFFu)      // tensor_dim0[31:16]
              | (1u << 16));                   // tensor_dim1 = 1 (low half)
  g1[3] = (int)((nelem & 0xFFFFu) << 16);      // tensor_dim1 hi=0 | tile_dim0=nelem
  g1[4] = (int)1u;                             // tile_dim1=1, tile_dim2=0 (unused)
  g1[5] = (int)nelem;                          // tensor_dim0_stride (low 32)
  g1[6] = 0;                                   // stride0 hi / stride1 lo
  g1[7] = 0;                                   // stride1 hi

  v4i g2 = {0, 0, 0, 0};                       // dims 2/3 unused (tile_dim2=0)
  v4i g3 = {0, 0, 0, 0};

#if defined(__clang_major__) && (__clang_major__ >= 23)
  v8i g4 = {0, 0, 0, 0, 0, 0, 0, 0};
  __builtin_amdgcn_tensor_load_to_lds(g0, g1, g2, g3, g4, 0);
#else
  __builtin_amdgcn_tensor_load_to_lds(g0, g1, g2, g3, 0);
#endif
}

__global__ __launch_bounds__(256)
void conditions_layer_kernel(const float* __restrict__ x,
                             const float* __restrict__ w1,
                             const float* __restrict__ w2,
                             float* __restrict__ out) {
  constexpr int D = 64;
  constexpr int U = 8192;
  constexpr int CHUNK = 1024;   // u-span per block; 256 threads * 4 f32

  __shared__ __align__(16) float shX[D];
  __shared__ __align__(16) float shW1[CHUNK];
  __shared__ __align__(16) float shW2[CHUNK];

  const int tid   = (int)threadIdx.x;
  const int b     = (int)blockIdx.y;
  const int ubase = (int)blockIdx.x * CHUNK;

  // Wave 0 issues the three TDM copies (TDM ignores EXEC; one issue per wave).
  if (tid == 0) {
    tdm_load_1d_f32(x + (size_t)b * D, (uint32_t)(uintptr_t)(void*)shX,  (uint32_t)D);
    tdm_load_1d_f32(w1 + ubase,        (uint32_t)(uintptr_t)(void*)shW1, (uint32_t)CHUNK);
    tdm_load_1d_f32(w2 + ubase,        (uint32_t)(uintptr_t)(void*)shW2, (uint32_t)CHUNK);
  }
  // Issuing wave drains TENSORcnt; others pass immediately. Barrier then
  // publishes the LDS tiles to all 8 waves.
  __builtin_amdgcn_s_wait_tensorcnt(0);
  __syncthreads();

  const int u = ubase + tid * 4;                 // 16B-aligned, no 64-wrap inside
  const bool greater = ((u >> 6) & 1) == 0;      // uniform across the 4 lanes' u's

  const v4f xv = *(const v4f*)&shX[(tid * 4) & (D - 1)];
  const v4f av = *(const v4f*)&shW1[tid * 4];
  const v4f bv = *(const v4f*)&shW2[tid * 4];

  v4f o;
#pragma unroll
  for (int j = 0; j < 4; ++j) {
    o[j] = greater ? fmaxf(xv[j] - av[j], 0.0f)
                   : fmaxf(bv[j] - xv[j], 0.0f);
  }

  // Streaming output: write-once, never re-read -> non-temporal b128 store.
  __builtin_nontemporal_store(o, (v4f*)(out + (size_t)b * U + u));
}

extern "C" void kernel_launch(void* const* d_in, const int* in_sizes, int n_in,
                              void* d_out, int out_size, void* d_ws, size_t ws_size,
                              hipStream_t stream) {
  (void)in_sizes; (void)n_in; (void)d_ws; (void)ws_size; (void)out_size;

  const float* x  = (const float*)d_in[0];   // (1024, 64)  f32
  const float* w1 = (const float*)d_in[1];   // (8192,)     f32
  const float* w2 = (const float*)d_in[2];   // (8192,)     f32
  float* out = (float*)d_out;                // (1024, 8192) f32

  dim3 grid(8192 / 1024, 1024);              // (u-chunks, batch rows)
  conditions_layer_kernel<<<grid, dim3(256), 0, stream>>>(x, w1, w2, out);
}